// BilinearInterpolation_66099546685972
// MI455X (gfx1250) — compile-verified
//
#include <hip/hip_runtime.h>

// Bilinear spatial-transformer sampling, MI455X (gfx1250).
// Bandwidth-bound gather: ~270 MB moved vs ~0.4 GFLOP (1.5 flop/byte) ->
// optimize data path, not math. WMMA deliberately unused: no matrix op here
// exceeds 6 FMAs/pixel. One thread per 4-channel output pixel; wave32 issues
// fully-coalesced 512 B B128 stores.

typedef float v4f __attribute__((ext_vector_type(4)));

#define B      32
#define IMG_H  512
#define IMG_W  512
#define OUT_H_ 512
#define OUT_W_ 512
#define NPIX   (OUT_H_ * OUT_W_)          // 262144 = 2^18 per batch
#define BLOCKS_PER_BATCH (NPIX / 256)     // 1024 = 2^10

__global__ __launch_bounds__(256) void bilinear_sampler_kernel(
    const float* __restrict__ image,      // [B, 512, 512, 4]
    const float* __restrict__ affine,     // [B, 6]
    float* __restrict__ out)              // [B, 512, 512, 4]
{
    const int blk = (int)blockIdx.x;
    // Batch index derived ONLY from blockIdx -> provably uniform -> the
    // affine coefficient fetch lowers to s_load (KMcnt scalar path).
    const int b   = blk >> 10;            // blk / BLOCKS_PER_BATCH
    const int gid = blk * 256 + (int)threadIdx.x;   // [0, B*NPIX)
    const int n   = gid & (NPIX - 1);
    const int iy  = n >> 9;
    const int ix  = n & (OUT_W_ - 1);

    const float* t = affine + b * 6;
    const float a0 = t[0], a1 = t[1], a2 = t[2];
    const float a3 = t[3], a4 = t[4], a5 = t[5];

    // linspace(-1, 1, 512): step = 2/511
    const float step = 2.0f / (float)(OUT_W_ - 1);
    const float xn = __builtin_fmaf((float)ix, step, -1.0f);
    const float yn = __builtin_fmaf((float)iy, step, -1.0f);

    // grid = A @ [xn, yn, 1]
    const float xg = __builtin_fmaf(a0, xn, __builtin_fmaf(a1, yn, a2));
    const float yg = __builtin_fmaf(a3, xn, __builtin_fmaf(a4, yn, a5));

    // to image coords: 0.5*(g+1)*W  (full W/H, no -1; matches reference)
    const float x = (xg + 1.0f) * (0.5f * (float)IMG_W);
    const float y = (yg + 1.0f) * (0.5f * (float)IMG_H);

    // truncation toward zero == astype(int32)
    int x0 = (int)x;
    int y0 = (int)y;
    int x1 = x0 + 1;
    int y1 = y0 + 1;
    x0 = min(max(x0, 0), IMG_W - 1);
    x1 = min(max(x1, 0), IMG_W - 1);
    y0 = min(max(y0, 0), IMG_H - 1);
    y1 = min(max(y1, 0), IMG_H - 1);

    // weights from CLIPPED corner coords (exactly as reference)
    const float xmf = (float)x0, xMf = (float)x1;
    const float ymf = (float)y0, yMf = (float)y1;
    const float dx0 = xMf - x, dx1 = x - xmf;
    const float dy0 = yMf - y, dy1 = y - ymf;
    const float wA = dx0 * dy0;   // (y0, x0)
    const float wB = dx0 * dy1;   // (y1, x0)
    const float wC = dx1 * dy0;   // (y0, x1)
    const float wD = dx1 * dy1;   // (y1, x1)

    // 16-byte gathers; input image is L2-resident (128 MiB < 192 MB L2),
    // default RT temporal hint keeps corner reuse in WGP$/L2.
    const v4f* __restrict__ img = (const v4f*)image;
    const long base = (long)b * (IMG_H * IMG_W);
    const long r0 = base + (long)(y0 * IMG_W);
    const long r1 = base + (long)(y1 * IMG_W);
    const v4f pA = img[r0 + x0];
    const v4f pB = img[r1 + x0];
    const v4f pC = img[r0 + x1];
    const v4f pD = img[r1 + x1];

    v4f r;
    r = pA * wA;
    r.x = __builtin_fmaf(pB.x, wB, r.x);
    r.y = __builtin_fmaf(pB.y, wB, r.y);
    r.z = __builtin_fmaf(pB.z, wB, r.z);
    r.w = __builtin_fmaf(pB.w, wB, r.w);
    r.x = __builtin_fmaf(pC.x, wC, r.x);
    r.y = __builtin_fmaf(pC.y, wC, r.y);
    r.z = __builtin_fmaf(pC.z, wC, r.z);
    r.w = __builtin_fmaf(pC.w, wC, r.w);
    r.x = __builtin_fmaf(pD.x, wD, r.x);
    r.y = __builtin_fmaf(pD.y, wD, r.y);
    r.z = __builtin_fmaf(pD.z, wD, r.z);
    r.w = __builtin_fmaf(pD.w, wD, r.w);

    // Streaming output: non-temporal B128 store so 128 MiB of results
    // doesn't evict the L2-resident source image.
    v4f* o = (v4f*)out;
    __builtin_nontemporal_store(r, &o[gid]);
}

extern "C" void kernel_launch(void* const* d_in, const int* in_sizes, int n_in,
                              void* d_out, int out_size, void* d_ws, size_t ws_size,
                              hipStream_t stream) {
    (void)in_sizes; (void)n_in; (void)d_ws; (void)ws_size; (void)out_size;
    const float* image  = (const float*)d_in[0];   // [32,512,512,4] fp32
    const float* affine = (const float*)d_in[1];   // [32,6] fp32
    float* out          = (float*)d_out;           // [32,512,512,4] fp32

    const int total  = B * NPIX;                   // 8,388,608 threads
    const int blocks = total / 256;                // 32768, exact
    bilinear_sampler_kernel<<<blocks, 256, 0, stream>>>(image, affine, out);
}